// FrequencyComplexModel_154618823029
// MI455X (gfx1250) — compile-verified
//
#include <hip/hip_runtime.h>
#include <math.h>

// ---------------------------------------------------------------------------
// FrequencyComplexModel for MI455X (gfx1250, wave32).
//
// Per row (4096 rows x 16384 samples):
//   pn[n]  = 0.01 * exclusive_prefix_sum(phase_steps)[n]
//   out[n] = clip(QD*floor((amp*cos(2pi*f0*n + ph + pn[n]) + 0.1*wn[n])/QD)+QD/2, -1, 1)
//
// Memory-bound (~768 MB @ 23.3 TB/s ~ 33 us floor). CDNA5 paths used:
//   * global_load_async_to_lds_b128/b32 (ASYNCcnt) to stage the step row in LDS
//   * v_wmma_f32_16x16x4_f32 x4 to compute the 256-wide block scan as
//     M(16x16) x UpperTriangularOnes(16x16)
// ---------------------------------------------------------------------------

#ifndef __has_builtin
#define __has_builtin(x) 0
#endif

#if __has_builtin(__builtin_amdgcn_global_load_async_to_lds_b128) && \
    __has_builtin(__builtin_amdgcn_global_load_async_to_lds_b32)
#define HAVE_ASYNC_LDS 1
#else
#define HAVE_ASYNC_LDS 0
#endif

#if __has_builtin(__builtin_amdgcn_s_wait_asynccnt)
#define WAIT_ASYNCCNT0() __builtin_amdgcn_s_wait_asynccnt(0)
#else
#define WAIT_ASYNCCNT0() asm volatile("s_wait_asynccnt 0" ::: "memory")
#endif

#if __has_builtin(__builtin_amdgcn_wmma_f32_16x16x4_f32)
#define HAVE_WMMA_F32X4 1
#else
#define HAVE_WMMA_F32X4 0
#endif

typedef float v2f __attribute__((ext_vector_type(2)));
typedef float v8f __attribute__((ext_vector_type(8)));
typedef int   v4i __attribute__((ext_vector_type(4)));

// Typed address-space pointers for the async-to-LDS builtins
// (b128 wants int4 AS1* / int4 AS3*, b32 wants int AS1* / int AS3*).
typedef __attribute__((address_space(1))) v4i* g_v4i_p;
typedef __attribute__((address_space(3))) v4i* l_v4i_p;
typedef __attribute__((address_space(1))) int* g_i32_p;
typedef __attribute__((address_space(3))) int* l_i32_p;

#define NSAMP 16384
#define NSTEP 16383            // N_SAMPLES - 1
#define TPB   256
#define CHUNK 64               // samples per thread
// LDS skew: +4 dwords per 64 dwords keeps 16B alignment and breaks the
// 32-way bank conflict of the per-thread contiguous serial scans (-> 2-way).
#define PAD_IDX(i) ((i) + 4 * ((i) >> 6))
#define LDS_STEPS (NSAMP + 4 * (NSAMP / 64))   // 17408 floats

__device__ __forceinline__ float quantize1(float x) {
    const float QD = 0.00784313725490196078f;  // 2/255 (matches f32(Q_DELTA))
    const float QH = 0.00392156862745098039f;  // 1/255
    float q = QD * floorf(x / QD) + QH;
    return fminf(fmaxf(q, -1.0f), 1.0f);
}

__global__ __launch_bounds__(TPB)
void freq_quant_kernel(const float* __restrict__ cond,
                       const float* __restrict__ wn,
                       const float* __restrict__ steps,
                       float* __restrict__ out) {
    __shared__ float s_steps[LDS_STEPS];  // raw steps, then in-place phase noise
    __shared__ float s_tot[TPB];          // per-thread chunk totals (scaled)
    __shared__ float s_scan[TPB];         // inclusive scan within 16-wide rows
    __shared__ float s_rowbase[16];       // exclusive scan of row totals

    const int t   = threadIdx.x;
    const int row = blockIdx.x;
    const float* __restrict__ srow = steps + (size_t)row * NSTEP;

    // ---- Stage phase_steps row into LDS (async DMA path on CDNA5) ----------
#if HAVE_ASYNC_LDS && defined(__AMDGCN__)
    #pragma unroll
    for (int k = 0; k < 16; ++k) {
        const int f = k * TPB + t;                    // float4 group id
        if (f < 4095) {                               // dwords [0, 16380)
            __builtin_amdgcn_global_load_async_to_lds_b128(
                (g_v4i_p)(srow + 4 * f),
                (l_v4i_p)&s_steps[4 * f + 4 * (f >> 4)],
                0, 0);
        }
    }
    if (t < 3) {                                      // tail dwords 16380..16382
        const int i = 16380 + t;
        __builtin_amdgcn_global_load_async_to_lds_b32(
            (g_i32_p)(srow + i),
            (l_i32_p)&s_steps[PAD_IDX(i)],
            0, 0);
    }
    if (t == 0) s_steps[PAD_IDX(16383)] = 0.0f;       // pad slot (step 16383 := 0)
    WAIT_ASYNCCNT0();
#else
    #pragma unroll
    for (int k = 0; k < 16; ++k) {
        const int f = k * TPB + t;
        if (f < 4095) {
            const float4 v = *(const float4*)(srow + 4 * f);
            *(float4*)&s_steps[4 * f + 4 * (f >> 4)] = v;
        }
    }
    if (t < 3) { const int i = 16380 + t; s_steps[PAD_IDX(i)] = srow[i]; }
    if (t == 0) s_steps[PAD_IDX(16383)] = 0.0f;
#endif
    __syncthreads();

    // ---- Phase A: per-thread chunk totals (scale each step like reference) -
    {
        float tot = 0.0f;
        const int base = t * (CHUNK + 4);   // == PAD_IDX(t*64)
        #pragma unroll 8
        for (int j = 0; j < CHUNK; ++j) tot = fmaf(0.01f, s_steps[base + j], tot);
        s_tot[t] = tot;
    }
    __syncthreads();

    // ---- Phase B: 256-wide scan of totals via WMMA: D = M(16x16) * U -------
    // M[r,c] = s_tot[16r+c]; U[k,n] = (n >= k).  Four K=4 accumulating WMMAs.
#if HAVE_WMMA_F32X4 && defined(__AMDGCN__)
    if (t < 32) {                       // wave 0, EXEC all ones inside branch
        const int half = t >> 4;        // VGPR-half select
        const int m    = t & 15;        // A row / B column
        v8f acc = {0.f, 0.f, 0.f, 0.f, 0.f, 0.f, 0.f, 0.f};
        #pragma unroll
        for (int s = 0; s < 4; ++s) {
            const int k0 = 4 * s + 2 * half;   // K indices held by this lane
            v2f a, b;
            a.x = s_tot[16 * m + k0];
            a.y = s_tot[16 * m + k0 + 1];
            b.x = (m >= k0)     ? 1.0f : 0.0f;
            b.y = (m >= k0 + 1) ? 1.0f : 0.0f;
            acc = __builtin_amdgcn_wmma_f32_16x16x4_f32(
                false, a, false, b, (short)0, acc, false, false);
        }
        #pragma unroll
        for (int v = 0; v < 8; ++v)     // D layout: VGPR v -> row v + 8*half
            s_scan[16 * (v + 8 * half) + m] = acc[v];
    }
#else
    if (t < 16) {
        float c = 0.0f;
        for (int j = 0; j < 16; ++j) { c += s_tot[16 * t + j]; s_scan[16 * t + j] = c; }
    }
#endif
    __syncthreads();

    // ---- Phase C: exclusive scan of the 16 row totals (tiny) ---------------
    if (t < 16) {
        float rb = 0.0f;
        for (int r = 0; r < t; ++r) rb += s_scan[16 * r + 15];
        s_rowbase[t] = rb;
    }
    __syncthreads();

    // ---- Phase D: in-place exclusive scan -> phase noise in LDS ------------
    {
        float cur = s_rowbase[t >> 4] + s_scan[t] - s_tot[t];  // exclusive base
        const int base = t * (CHUNK + 4);
        #pragma unroll 8
        for (int j = 0; j < CHUNK; ++j) {
            const float sv = s_steps[base + j];
            s_steps[base + j] = cur;                 // pn for sample t*64+j
            cur = fmaf(0.01f, sv, cur);
        }
    }
    __syncthreads();

    // ---- Phase E: coalesced float4 streaming compute -----------------------
    const float amp = cond[3 * row + 0];
    const float f0  = cond[3 * row + 1];
    const float ph  = cond[3 * row + 2];
    const float w   = 6.2831853071795864769f * f0;   // (2*pi)*f0 like reference
    const float* __restrict__ wrow = wn  + (size_t)row * NSAMP;
    float* __restrict__       orow = out + (size_t)row * NSAMP;

    #pragma unroll 4
    for (int c = 0; c < 16; ++c) {
        const int n0 = c * 1024 + t * 4;
        const int li = c * 1088 + t * 4 + 4 * (t >> 4);   // == PAD_IDX(n0)
        const float4 pn = *(const float4*)&s_steps[li];
        const float4 g  = *(const float4*)(wrow + n0);
        float4 o;
        o.x = quantize1(fmaf(amp, cosf(w * (float)(n0 + 0) + ph + pn.x), 0.1f * g.x));
        o.y = quantize1(fmaf(amp, cosf(w * (float)(n0 + 1) + ph + pn.y), 0.1f * g.y));
        o.z = quantize1(fmaf(amp, cosf(w * (float)(n0 + 2) + ph + pn.z), 0.1f * g.z));
        o.w = quantize1(fmaf(amp, cosf(w * (float)(n0 + 3) + ph + pn.w), 0.1f * g.w));
        *(float4*)(orow + n0) = o;
    }
}

extern "C" void kernel_launch(void* const* d_in, const int* in_sizes, int n_in,
                              void* d_out, int out_size, void* d_ws, size_t ws_size,
                              hipStream_t stream) {
    const float* cond   = (const float*)d_in[0];  // (B, 3)
    const float* wnoise = (const float*)d_in[1];  // (B, 16384)
    const float* steps  = (const float*)d_in[2];  // (B, 16383)
    float* out = (float*)d_out;                   // (B, 16384) float32
    const int batch = in_sizes[0] / 3;            // 4096
    freq_quant_kernel<<<dim3(batch), dim3(TPB), 0, stream>>>(cond, wnoise, steps, out);
}